// GCN_Encoder_26268019982946
// MI455X (gfx1250) — compile-verified
//
#include <hip/hip_runtime.h>
#include <math.h>

typedef __attribute__((ext_vector_type(2))) float v2f;
typedef __attribute__((ext_vector_type(8))) float v8f;

// ---------------------------------------------------------------------------
// Degree / normalization kernels
// ---------------------------------------------------------------------------
__global__ void fill_ones(float* __restrict__ p, int n) {
  int i = blockIdx.x * blockDim.x + threadIdx.x;
  if (i < n) p[i] = 1.0f;  // self-loop weight contributes 1 to every degree
}

__global__ void deg_scatter(const int* __restrict__ dst, const float* __restrict__ ew,
                            float* __restrict__ deg, int E) {
  int e = blockIdx.x * blockDim.x + threadIdx.x;
  if (e < E) atomicAdd(&deg[dst[e]], ew[e]);
}

__global__ void deg_to_dinv(float* __restrict__ d, int n) {
  int i = blockIdx.x * blockDim.x + threadIdx.x;
  if (i < n) {
    float v = d[i];
    d[i] = (v > 0.0f) ? rsqrtf(v) : 0.0f;
  }
}

// ---------------------------------------------------------------------------
// Register-blocked WMMA GEMM (fast path, compile-time K and N):
//   - one wave computes a 16 x (NT*16) output strip: A pair loaded once per
//     k-step, reused across NT back-to-back v_wmma_f32_16x16x4_f32.
//   - B staged in LDS once per 256-thread block (8 m-tiles share it).
//   - LDS row stride padded by +16 floats: half-wave row offset of 2 rows
//     = 2*(N+16) floats = +32 banks -> lanes 0-15 / 16-31 use disjoint banks.
// A layout (16x4 f32): lanes 0-15 hold M=lane, K={0,1}; lanes 16-31 K={2,3}.
// B layout (4x16 f32): lanes 0-15 hold N=lane, K={0,1}; lanes 16-31 K={2,3}.
// C/D layout: VGPR r -> M=r (lanes 0-15) / M=r+8 (lanes 16-31), N=lane&15.
// ---------------------------------------------------------------------------
template <int KK, int NT>
__global__ void __launch_bounds__(256)
gemm_wmma_f32_rb(const float* __restrict__ A, const float* __restrict__ B,
                 float* __restrict__ C, int M) {
  constexpr int N   = NT * 16;
  constexpr int LDB = N + 16;              // padded LDS stride (bank shift)
  __shared__ float ldsB[KK * LDB];

  // Cooperative B stage (ALL waves, before any wave may exit).
  const int tid = threadIdx.y * 32 + threadIdx.x;
  for (int i = tid; i < KK * N; i += 256) {
    int r = i / N;
    int c = i - r * N;
    ldsB[r * LDB + c] = B[(size_t)r * N + c];
  }
  __syncthreads();

  const int row0 = (blockIdx.x * 8 + threadIdx.y) * 16;
  if (row0 + 16 > M) return;               // wave-uniform: EXEC stays all-1s

  const int lane = threadIdx.x;
  const int half = lane >> 4;              // 0: K pair {0,1}, 1: K pair {2,3}
  const int idx  = lane & 15;              // M index for A, N index for B/C

  const float* arow = A + (size_t)(row0 + idx) * KK + 2 * half;
  v8f acc[NT];
#pragma unroll
  for (int t = 0; t < NT; ++t) acc[t] = v8f{};

  for (int k = 0; k < KK; k += 4) {
    v2f a;
    a.x = arow[k];
    a.y = arow[k + 1];
    const float* bp = ldsB + (k + 2 * half) * LDB + idx;
#pragma unroll
    for (int t = 0; t < NT; ++t) {
      v2f b;
      b.x = bp[16 * t];
      b.y = bp[16 * t + LDB];
      acc[t] = __builtin_amdgcn_wmma_f32_16x16x4_f32(false, a, false, b,
                                                     (short)0, acc[t], false, false);
    }
  }

  float* crow = C + (size_t)(row0 + 8 * half) * N + idx;
#pragma unroll
  for (int t = 0; t < NT; ++t)
#pragma unroll
    for (int r = 0; r < 8; ++r)
      crow[(size_t)r * N + 16 * t] = acc[t][r];
}

// ---------------------------------------------------------------------------
// Generic WMMA GEMM fallback (runtime dims, one wave per 16x16 tile).
// ---------------------------------------------------------------------------
__global__ void __launch_bounds__(256)
gemm16x16_wmma_f32(const float* __restrict__ A, const float* __restrict__ B,
                   float* __restrict__ C, int M, int K, int N) {
  const int lane = threadIdx.x;
  const int row0 = blockIdx.x * 16;
  const int col0 = threadIdx.y * 16;
  if (row0 + 16 > M) return;               // wave-uniform
  const int half = lane >> 4;
  const int idx  = lane & 15;

  const float* arow = A + (size_t)(row0 + idx) * K + 2 * half;
  v8f acc = {};
#pragma unroll 4
  for (int k = 0; k < K; k += 4) {
    v2f a, b;
    a.x = arow[k];
    a.y = arow[k + 1];
    const float* bp = B + (size_t)(k + 2 * half) * N + col0 + idx;
    b.x = bp[0];
    b.y = bp[N];
    acc = __builtin_amdgcn_wmma_f32_16x16x4_f32(false, a, false, b,
                                                (short)0, acc, false, false);
  }
  float* crow = C + (size_t)(row0 + 8 * half) * N + col0 + idx;
#pragma unroll
  for (int r = 0; r < 8; ++r) crow[(size_t)r * N] = acc[r];
}

// ---------------------------------------------------------------------------
// agg[i,:] = bias + dinv[i]^2 * h[i,:]   (self-loop term + bias, float4 wide)
// ---------------------------------------------------------------------------
__global__ void self_loop_bias(const float* __restrict__ h, const float* __restrict__ dinv,
                               const float* __restrict__ bias, float* __restrict__ out,
                               int n, int F) {
  int t = blockIdx.x * blockDim.x + threadIdx.x;  // over n * F/4
  int q = F >> 2;
  int i = t / q;
  if (i >= n) return;
  int f = (t - i * q) << 2;
  float di = dinv[i];
  float s = di * di;
  const float4 hv = *(const float4*)(h + (size_t)i * F + f);
  float4 o;
  o.x = bias[f + 0] + s * hv.x;
  o.y = bias[f + 1] + s * hv.y;
  o.z = bias[f + 2] + s * hv.z;
  o.w = bias[f + 3] + s * hv.w;
  *(float4*)(out + (size_t)i * F + f) = o;
}

// ---------------------------------------------------------------------------
// Edge scatter: one wave per edge (32 lanes x float4 = 128 feats).
// out[dst] += h[src] * (dinv[src]*ew*dinv[dst])  via global_atomic_add_f32.
// ---------------------------------------------------------------------------
__global__ void edge_scatter(const int* __restrict__ src, const int* __restrict__ dst,
                             const float* __restrict__ ew, const float* __restrict__ dinv,
                             const float* __restrict__ h, float* __restrict__ out,
                             int E, int F) {
  int t = blockIdx.x * blockDim.x + threadIdx.x;
  int lpe = F >> 2;                        // lanes per edge (32 for F=128)
  int e = t / lpe;
  if (e >= E) return;
  int f = (t - e * lpe) << 2;
  int s = src[e], d = dst[e];
  float norm = dinv[s] * ew[e] * dinv[d];
  const float4 hv = *(const float4*)(h + (size_t)s * F + f);
  float* op = out + (size_t)d * F + f;
  atomicAdd(op + 0, hv.x * norm);
  atomicAdd(op + 1, hv.y * norm);
  atomicAdd(op + 2, hv.z * norm);
  atomicAdd(op + 3, hv.w * norm);
}

__global__ void relu_inplace(float* __restrict__ p, int total4) {
  int t = blockIdx.x * blockDim.x + threadIdx.x;
  if (t >= total4) return;
  float4 v = *(float4*)(p + (size_t)t * 4);
  v.x = fmaxf(v.x, 0.0f);
  v.y = fmaxf(v.y, 0.0f);
  v.z = fmaxf(v.z, 0.0f);
  v.w = fmaxf(v.w, 0.0f);
  *(float4*)(p + (size_t)t * 4) = v;
}

// ---------------------------------------------------------------------------
// log_softmax over 64 classes: one wave per row, 2 columns per lane.
// Adds FC bias here (GEMM produced raw logits).
// ---------------------------------------------------------------------------
__global__ void logsoftmax64(const float* __restrict__ logits, const float* __restrict__ fcb,
                             float* __restrict__ out, int n, int C) {
  int wavesPerBlock = blockDim.x >> 5;
  int row = blockIdx.x * wavesPerBlock + (threadIdx.x >> 5);
  int lane = threadIdx.x & 31;
  if (row >= n) return;
  const float* lr = logits + (size_t)row * C;
  float v0 = lr[lane] + fcb[lane];
  float v1 = lr[lane + 32] + fcb[lane + 32];
  float m = fmaxf(v0, v1);
#pragma unroll
  for (int off = 16; off > 0; off >>= 1) m = fmaxf(m, __shfl_xor(m, off, 32));
  float ssum = __expf(v0 - m) + __expf(v1 - m);
#pragma unroll
  for (int off = 16; off > 0; off >>= 1) ssum += __shfl_xor(ssum, off, 32);
  float lse = m + __logf(ssum);
  float* orow = out + (size_t)row * C;
  orow[lane] = v0 - lse;
  orow[lane + 32] = v1 - lse;
}

// ---------------------------------------------------------------------------
extern "C" void kernel_launch(void* const* d_in, const int* in_sizes, int n_in,
                              void* d_out, int out_size, void* d_ws, size_t ws_size,
                              hipStream_t stream) {
  const float* x    = (const float*)d_in[0];
  const int*   ei   = (const int*)d_in[1];
  const float* ew   = (const float*)d_in[2];
  const float* W1   = (const float*)d_in[3];
  const float* b1   = (const float*)d_in[4];
  const float* W2   = (const float*)d_in[5];
  const float* b2   = (const float*)d_in[6];
  const float* fcW  = (const float*)d_in[7];
  const float* fcb  = (const float*)d_in[8];
  float* out = (float*)d_out;

  const int nclass = in_sizes[8];                 // 64
  const int nhid   = in_sizes[4];                 // 128
  const int nfeat  = in_sizes[3] / nhid;          // 128
  const int E      = in_sizes[1] / 2;             // 800000
  const int n      = in_sizes[0] / nfeat;         // 50000
  const int* src = ei;
  const int* dst = ei + E;

  // Workspace: dinv [n] | hbuf [n*nhid] | agg [n*nhid]
  float* dinv = (float*)d_ws;
  float* hbuf = dinv + (size_t)((n + 255) & ~255);
  float* agg  = hbuf + (size_t)n * nhid;

  const int T = 256;
  // --- degrees -> dinv ---
  fill_ones<<<(n + T - 1) / T, T, 0, stream>>>(dinv, n);
  deg_scatter<<<(E + T - 1) / T, T, 0, stream>>>(dst, ew, dinv, E);
  deg_to_dinv<<<(n + T - 1) / T, T, 0, stream>>>(dinv, n);

  const int mtiles = (n + 15) / 16;
  const int nf4 = n * (nhid >> 2);
  const int ef  = E * (nhid >> 2);
  const bool fast = (nfeat == 128 && nhid == 128 && nclass == 64);
  const int rbBlocks = (mtiles + 7) / 8;

  // --- layer 1: h = x @ W1 ---
  if (fast)
    gemm_wmma_f32_rb<128, 8><<<rbBlocks, dim3(32, 8), 0, stream>>>(x, W1, hbuf, n);
  else
    gemm16x16_wmma_f32<<<dim3(mtiles), dim3(32, nhid / 16), 0, stream>>>(
        x, W1, hbuf, n, nfeat, nhid);
  self_loop_bias<<<(nf4 + T - 1) / T, T, 0, stream>>>(hbuf, dinv, b1, agg, n, nhid);
  edge_scatter<<<(ef + T - 1) / T, T, 0, stream>>>(src, dst, ew, dinv, hbuf, agg, E, nhid);
  relu_inplace<<<(nf4 + T - 1) / T, T, 0, stream>>>(agg, nf4);

  // --- layer 2: h = agg @ W2 ---
  if (fast)
    gemm_wmma_f32_rb<128, 8><<<rbBlocks, dim3(32, 8), 0, stream>>>(agg, W2, hbuf, n);
  else
    gemm16x16_wmma_f32<<<dim3(mtiles), dim3(32, nhid / 16), 0, stream>>>(
        agg, W2, hbuf, n, nhid, nhid);
  self_loop_bias<<<(nf4 + T - 1) / T, T, 0, stream>>>(hbuf, dinv, b2, agg, n, nhid);
  edge_scatter<<<(ef + T - 1) / T, T, 0, stream>>>(src, dst, ew, dinv, hbuf, agg, E, nhid);
  relu_inplace<<<(nf4 + T - 1) / T, T, 0, stream>>>(agg, nf4);

  // --- FC: logits = agg @ fcW ---
  if (fast)
    gemm_wmma_f32_rb<128, 4><<<rbBlocks, dim3(32, 8), 0, stream>>>(agg, fcW, hbuf, n);
  else
    gemm16x16_wmma_f32<<<dim3(mtiles), dim3(32, nclass / 16), 0, stream>>>(
        agg, fcW, hbuf, n, nhid, nclass);

  // --- log_softmax -> out ---
  int rowsPerBlock = T / 32;
  logsoftmax64<<<(n + rowsPerBlock - 1) / rowsPerBlock, T, 0, stream>>>(
      hbuf, fcb, out, n, nclass);
}